// Flow_Attention_111669150274
// MI455X (gfx1250) — compile-verified
//
#include <hip/hip_runtime.h>
#include <math.h>

typedef float v2f __attribute__((ext_vector_type(2)));
typedef float v8f __attribute__((ext_vector_type(8)));

#define H_HEADS 16
#define HD_DIM  64
#define EPSF    1e-6f

__device__ __forceinline__ float sigmoidf_(float x) {
    return 1.0f / (1.0f + __expf(-x));
}

// ---------------------------------------------------------------------------
// Generic GEMM: C[m,n] = act( sum_k A[m,k] * W[n,k] + bias[n] )
// A: [Mrows x Kdim] row-major, W: [Ncols x Kdim] row-major (acts as B^T).
// out_mode 0: C row-major [Mrows x Ncols]
// out_mode 1: C in [B,H,L,HD] layout (row -> (b,l), col -> (h,d))
// Uses V_WMMA_F32_16X16X4_F32; block computes a 64x64 tile (8 waves, each
// wave owns one 16-row strip and two 16-col tiles).
// ---------------------------------------------------------------------------
__global__ __launch_bounds__(256) void gemm_bias_act(
    const float* __restrict__ A,
    const float* __restrict__ W,
    const float* __restrict__ bias,
    float* __restrict__ C,
    int Mrows, int Kdim, int Ncols,
    int act, int out_mode, int Lseq)
{
    __shared__ float As[64][36];   // pad 36: gcd(36,64)=4 -> conflict-free row-varying reads
    __shared__ float Ws[64][36];

    const int tid  = threadIdx.x;
    const int lane = tid & 31;
    const int wid  = tid >> 5;
    const int hv   = lane >> 4;    // half of wave (0/1)
    const int lm   = lane & 15;
    const int m0   = (wid & 3) * 16;
    const int n0   = (wid >> 2) * 32;
    const int rowBase = blockIdx.y * 64;
    const int colBase = blockIdx.x * 64;

    v8f acc0 = {0.f,0.f,0.f,0.f,0.f,0.f,0.f,0.f};
    v8f acc1 = {0.f,0.f,0.f,0.f,0.f,0.f,0.f,0.f};

    for (int k0 = 0; k0 < Kdim; k0 += 32) {
        // stage 64x32 tiles of A and W (512 float4 each, 2 per thread)
        #pragma unroll
        for (int i = 0; i < 2; ++i) {
            int idx = tid * 2 + i;          // 0..511
            int r   = idx >> 3;             // 0..63
            int q4  = idx & 7;              // float4 slot in 32-wide tile
            float4 av = *(reinterpret_cast<const float4*>(
                              A + (size_t)(rowBase + r) * Kdim + k0) + q4);
            As[r][q4*4+0] = av.x; As[r][q4*4+1] = av.y;
            As[r][q4*4+2] = av.z; As[r][q4*4+3] = av.w;
            float4 wv = *(reinterpret_cast<const float4*>(
                              W + (size_t)(colBase + r) * Kdim + k0) + q4);
            Ws[r][q4*4+0] = wv.x; Ws[r][q4*4+1] = wv.y;
            Ws[r][q4*4+2] = wv.z; Ws[r][q4*4+3] = wv.w;
        }
        __syncthreads();
        #pragma unroll
        for (int kk = 0; kk < 8; ++kk) {
            int kb = kk*4 + hv*2;
            v2f a;  a[0]  = As[m0+lm][kb];      a[1]  = As[m0+lm][kb+1];
            v2f b0; b0[0] = Ws[n0+lm][kb];      b0[1] = Ws[n0+lm][kb+1];
            v2f b1; b1[0] = Ws[n0+16+lm][kb];   b1[1] = Ws[n0+16+lm][kb+1];
            acc0 = __builtin_amdgcn_wmma_f32_16x16x4_f32(
                       false, a, false, b0, (short)0, acc0, false, false);
            acc1 = __builtin_amdgcn_wmma_f32_16x16x4_f32(
                       false, a, false, b1, (short)0, acc1, false, false);
        }
        __syncthreads();
    }

    const int col0 = colBase + n0 + lm;
    const int col1 = col0 + 16;
    const float bi0 = bias[col0];
    const float bi1 = bias[col1];
    #pragma unroll
    for (int vv = 0; vv < 8; ++vv) {
        int gr = rowBase + m0 + vv + hv*8;     // C layout: m = vv + half*8
        float v0 = acc0[vv] + bi0;
        float v1 = acc1[vv] + bi1;
        if (act) { v0 = sigmoidf_(v0); v1 = sigmoidf_(v1); }
        if (out_mode == 0) {
            C[(size_t)gr * Ncols + col0] = v0;
            C[(size_t)gr * Ncols + col1] = v1;
        } else {
            int b_ = gr / Lseq, li = gr % Lseq;
            int h0 = col0 / HD_DIM, d0 = col0 % HD_DIM;
            int h1 = col1 / HD_DIM, d1 = col1 % HD_DIM;
            C[(((size_t)b_*H_HEADS + h0)*Lseq + li)*HD_DIM + d0] = v0;
            C[(((size_t)b_*H_HEADS + h1)*Lseq + li)*HD_DIM + d1] = v1;
        }
    }
}

// out[bh,d] = sum_l X[bh,l,d] * (wgt ? wgt[bh,l] : 1)
__global__ __launch_bounds__(256) void colsum_weighted(
    const float* __restrict__ X, const float* __restrict__ wgt,
    float* __restrict__ out, int L)
{
    __shared__ float red[256];
    const int bh  = blockIdx.x;
    const int d   = threadIdx.x & 63;
    const int seg = threadIdx.x >> 6;
    float s = 0.f;
    for (int l = seg; l < L; l += 4) {
        float w = wgt ? wgt[(size_t)bh*L + l] : 1.0f;
        s += X[((size_t)bh*L + l)*64 + d] * w;
    }
    red[threadIdx.x] = s;
    __syncthreads();
    if (threadIdx.x < 64)
        out[(size_t)bh*64 + d] = red[d] + red[64+d] + red[128+d] + red[192+d];
}

// sink_inc = 1/dot(q+eps, ksum+eps) ; src_out = 1/dot(k+eps, qsum+eps)
__global__ __launch_bounds__(256) void flow_recip(
    const float* __restrict__ q, const float* __restrict__ k,
    const float* __restrict__ qsum, const float* __restrict__ ksum,
    float* __restrict__ sink_inc, float* __restrict__ src_out, int L)
{
    size_t gid = (size_t)blockIdx.x * 256 + threadIdx.x;   // over BH*L
    size_t bh  = gid / (size_t)L;
    const float* qp = q + gid*64;
    const float* kp = k + gid*64;
    const float* ks = ksum + bh*64;
    const float* qs = qsum + bh*64;
    float s1 = 0.f, s2 = 0.f;
    #pragma unroll 8
    for (int d = 0; d < 64; ++d) {
        s1 += (qp[d] + EPSF) * (ks[d] + EPSF);
        s2 += (kp[d] + EPSF) * (qs[d] + EPSF);
    }
    sink_inc[gid] = 1.0f / s1;
    src_out[gid]  = 1.0f / s2;
}

// sink_alloc = sigmoid(dot(q+eps, t1+eps))  (L/s == 1)
// cs_src     = clip(dot(k+eps, t2+eps), -1, 1)
__global__ __launch_bounds__(256) void flow_conserved(
    const float* __restrict__ q, const float* __restrict__ k,
    const float* __restrict__ t1, const float* __restrict__ t2,
    float* __restrict__ sink_alloc, float* __restrict__ cs_src, int L)
{
    size_t gid = (size_t)blockIdx.x * 256 + threadIdx.x;
    size_t bh  = gid / (size_t)L;
    const float* qp = q + gid*64;
    const float* kp = k + gid*64;
    const float* a1 = t1 + bh*64;
    const float* a2 = t2 + bh*64;
    float s1 = 0.f, s2 = 0.f;
    #pragma unroll 8
    for (int d = 0; d < 64; ++d) {
        s1 += (qp[d] + EPSF) * (a1[d] + EPSF);
        s2 += (kp[d] + EPSF) * (a2[d] + EPSF);
    }
    sink_alloc[gid] = sigmoidf_(s1);
    cs_src[gid]     = fminf(1.0f, fmaxf(-1.0f, s2));
}

// comp[bh,l] = softmax_l(cs[bh,:]) * scale
__global__ __launch_bounds__(256) void softmax_scale(
    const float* __restrict__ cs, float* __restrict__ comp, int L, float scale)
{
    __shared__ float red[256];
    const int bh = blockIdx.x;
    const int t  = threadIdx.x;
    const float* x = cs + (size_t)bh*L;
    float m = -1e30f;
    for (int l = t; l < L; l += 256) m = fmaxf(m, x[l]);
    red[t] = m; __syncthreads();
    for (int s = 128; s > 0; s >>= 1) {
        if (t < s) red[t] = fmaxf(red[t], red[t+s]);
        __syncthreads();
    }
    m = red[0]; __syncthreads();
    float sum = 0.f;
    for (int l = t; l < L; l += 256) sum += __expf(x[l] - m);
    red[t] = sum; __syncthreads();
    for (int s = 128; s > 0; s >>= 1) {
        if (t < s) red[t] += red[t+s];
        __syncthreads();
    }
    float inv = scale / red[0];
    for (int l = t; l < L; l += 256) comp[(size_t)bh*L + l] = __expf(x[l] - m) * inv;
}

// kv[bh,d,m] = sum_l k[bh,l,d] * v[bh,l,m] * comp[bh,l]   (64x64xL WMMA GEMM)
__global__ __launch_bounds__(256) void kv_gemm(
    const float* __restrict__ kmat, const float* __restrict__ vmat,
    const float* __restrict__ comp, float* __restrict__ kv, int L)
{
    __shared__ float Ks[32][80];   // pad 80: two wave-halves hit disjoint bank ranges
    __shared__ float Vs[32][80];
    const int bh   = blockIdx.x;
    const int tid  = threadIdx.x;
    const int lane = tid & 31;
    const int wid  = tid >> 5;
    const int hv   = lane >> 4;
    const int lm   = lane & 15;
    const int d0   = (wid & 3) * 16;
    const int n0   = (wid >> 2) * 32;

    v8f acc0 = {0.f,0.f,0.f,0.f,0.f,0.f,0.f,0.f};
    v8f acc1 = {0.f,0.f,0.f,0.f,0.f,0.f,0.f,0.f};

    for (int l0 = 0; l0 < L; l0 += 32) {
        #pragma unroll
        for (int i = 0; i < 8; ++i) {
            int idx = tid + i*256;           // 0..2047
            int ll  = idx >> 6;
            int dd  = idx & 63;
            size_t g = ((size_t)bh*L + l0 + ll)*64 + dd;
            Ks[ll][dd] = kmat[g];
            Vs[ll][dd] = vmat[g] * comp[(size_t)bh*L + l0 + ll];
        }
        __syncthreads();
        #pragma unroll
        for (int kk = 0; kk < 8; ++kk) {
            int kb = kk*4 + hv*2;
            v2f a;  a[0]  = Ks[kb][d0+lm];    a[1]  = Ks[kb+1][d0+lm];
            v2f b0; b0[0] = Vs[kb][n0+lm];    b0[1] = Vs[kb+1][n0+lm];
            v2f b1; b1[0] = Vs[kb][n0+16+lm]; b1[1] = Vs[kb+1][n0+16+lm];
            acc0 = __builtin_amdgcn_wmma_f32_16x16x4_f32(
                       false, a, false, b0, (short)0, acc0, false, false);
            acc1 = __builtin_amdgcn_wmma_f32_16x16x4_f32(
                       false, a, false, b1, (short)0, acc1, false, false);
        }
        __syncthreads();
    }
    #pragma unroll
    for (int vv = 0; vv < 8; ++vv) {
        int dd = d0 + vv + hv*8;
        kv[(size_t)bh*4096 + dd*64 + n0 + lm]      = acc0[vv];
        kv[(size_t)bh*4096 + dd*64 + n0 + 16 + lm] = acc1[vv];
    }
}

// attn[b,l,h*64+m] = sink_alloc[bh,l] * sum_d (q[bh,l,d]*sink_inc[bh,l]) * kv[bh,d,m]
__global__ __launch_bounds__(256) void out_gemm(
    const float* __restrict__ qmat, const float* __restrict__ kv,
    const float* __restrict__ sink_inc, const float* __restrict__ sink_alloc,
    float* __restrict__ attn, int L)
{
    __shared__ float Qs[64][68];   // row-varying lane reads -> pad 68
    __shared__ float KVs[64][80];  // row-fixed lane reads   -> pad 80
    const int bh  = blockIdx.y;
    const int l0  = blockIdx.x * 64;
    const int tid = threadIdx.x;
    const int lane = tid & 31;
    const int wid  = tid >> 5;
    const int hv   = lane >> 4;
    const int lm   = lane & 15;
    const int m0   = (wid & 3) * 16;
    const int n0   = (wid >> 2) * 32;

    #pragma unroll
    for (int i = 0; i < 16; ++i) {
        int idx = tid + i*256;               // 0..4095
        int ll  = idx >> 6;
        int dd  = idx & 63;
        Qs[ll][dd] = qmat[((size_t)bh*L + l0 + ll)*64 + dd]
                   * sink_inc[(size_t)bh*L + l0 + ll];
        KVs[idx >> 6][idx & 63] = kv[(size_t)bh*4096 + idx];
    }
    __syncthreads();

    v8f acc0 = {0.f,0.f,0.f,0.f,0.f,0.f,0.f,0.f};
    v8f acc1 = {0.f,0.f,0.f,0.f,0.f,0.f,0.f,0.f};
    #pragma unroll
    for (int kk = 0; kk < 16; ++kk) {
        int kb = kk*4 + hv*2;
        v2f a;  a[0]  = Qs[m0+lm][kb];       a[1]  = Qs[m0+lm][kb+1];
        v2f b0; b0[0] = KVs[kb][n0+lm];      b0[1] = KVs[kb+1][n0+lm];
        v2f b1; b1[0] = KVs[kb][n0+16+lm];   b1[1] = KVs[kb+1][n0+16+lm];
        acc0 = __builtin_amdgcn_wmma_f32_16x16x4_f32(
                   false, a, false, b0, (short)0, acc0, false, false);
        acc1 = __builtin_amdgcn_wmma_f32_16x16x4_f32(
                   false, a, false, b1, (short)0, acc1, false, false);
    }

    const int b_ = bh / H_HEADS;
    const int h  = bh % H_HEADS;
    #pragma unroll
    for (int vv = 0; vv < 8; ++vv) {
        int li = l0 + m0 + vv + hv*8;
        float sa = sink_alloc[(size_t)bh*L + li];
        attn[((size_t)b_*L + li)*1024 + h*64 + n0 + lm]      = acc0[vv] * sa;
        attn[((size_t)b_*L + li)*1024 + h*64 + n0 + 16 + lm] = acc1[vv] * sa;
    }
}

extern "C" void kernel_launch(void* const* d_in, const int* in_sizes, int n_in,
                              void* d_out, int out_size, void* d_ws, size_t ws_size,
                              hipStream_t stream) {
    (void)in_sizes; (void)n_in; (void)out_size; (void)ws_size;
    const float* x  = (const float*)d_in[0];
    const float* Wq = (const float*)d_in[1];
    const float* bq = (const float*)d_in[2];
    const float* Wk = (const float*)d_in[3];
    const float* bk = (const float*)d_in[4];
    const float* Wv = (const float*)d_in[5];
    const float* bv = (const float*)d_in[6];
    const float* Wo = (const float*)d_in[7];
    const float* bo = (const float*)d_in[8];
    float* out = (float*)d_out;

    const int B = 4, L = 2048, D = 1024, BH = B * H_HEADS;
    const size_t NQ = (size_t)B * L * D;          // 8,388,608 floats

    float* ws = (float*)d_ws;
    float* q          = ws;
    float* kbuf       = q + NQ;
    float* vbuf       = kbuf + NQ;
    float* attn       = vbuf + NQ;
    float* kvb        = attn + NQ;                // BH*64*64
    float* qsum       = kvb + (size_t)BH*64*64;
    float* ksum       = qsum + (size_t)BH*64;
    float* t1         = ksum + (size_t)BH*64;
    float* t2         = t1   + (size_t)BH*64;
    float* sink_inc   = t2   + (size_t)BH*64;
    float* src_out    = sink_inc   + (size_t)BH*L;
    float* sink_alloc = src_out    + (size_t)BH*L;
    float* cs_src     = sink_alloc + (size_t)BH*L;
    float* comp       = cs_src     + (size_t)BH*L;

    dim3 blk(256);
    dim3 gGemm(D/64, (B*L)/64);                   // 16 x 128

    // Q = sigmoid(x Wq^T + bq), K = sigmoid(...), V = x Wv^T + bv   (BHLD layout)
    gemm_bias_act<<<gGemm, blk, 0, stream>>>(x, Wq, bq, q,    B*L, D, D, 1, 1, L);
    gemm_bias_act<<<gGemm, blk, 0, stream>>>(x, Wk, bk, kbuf, B*L, D, D, 1, 1, L);
    gemm_bias_act<<<gGemm, blk, 0, stream>>>(x, Wv, bv, vbuf, B*L, D, D, 0, 1, L);

    colsum_weighted<<<BH, blk, 0, stream>>>(q,    nullptr, qsum, L);
    colsum_weighted<<<BH, blk, 0, stream>>>(kbuf, nullptr, ksum, L);

    flow_recip<<<(BH*L)/256, blk, 0, stream>>>(q, kbuf, qsum, ksum, sink_inc, src_out, L);

    colsum_weighted<<<BH, blk, 0, stream>>>(kbuf, src_out,  t1, L);
    colsum_weighted<<<BH, blk, 0, stream>>>(q,    sink_inc, t2, L);

    flow_conserved<<<(BH*L)/256, blk, 0, stream>>>(q, kbuf, t1, t2, sink_alloc, cs_src, L);
    softmax_scale<<<BH, blk, 0, stream>>>(cs_src, comp, L, (float)L);

    kv_gemm<<<BH, blk, 0, stream>>>(kbuf, vbuf, comp, kvb, L);

    dim3 gOut(L/64, BH);                          // 32 x 64
    out_gemm<<<gOut, blk, 0, stream>>>(q, kvb, sink_inc, sink_alloc, attn, L);

    // final projection: out = attn Wo^T + bo  (row-major [B*L x D])
    gemm_bias_act<<<gGemm, blk, 0, stream>>>(attn, Wo, bo, out, B*L, D, D, 0, 0, L);
}